// NodeThenActionPolicy_69114613730577
// MI455X (gfx1250) — compile-verified
//
#include <hip/hip_runtime.h>
#include <hip/hip_bf16.h>
#include <math.h>

// Problem constants (match reference)
#define G    512
#define NPG  512
#define D    128
#define A    16
#define NEGV (-1000000000.0f)

#define NCOL  48      // 34 used columns padded to 3 N-tiles of 16
#define LSTR  49      // LDS logits row stride (odd -> conflict free)
#define PHN   256     // nodes per phase

typedef __attribute__((ext_vector_type(2))) float v2f;
typedef __attribute__((ext_vector_type(8))) float v8f;

__global__ __launch_bounds__(256) void
fused_policy_kernel(const int* __restrict__ a_sel,
                    const float* __restrict__ h,
                    const unsigned char* __restrict__ amask,
                    const float* __restrict__ w_node,
                    const float* __restrict__ w_an,
                    const float* __restrict__ b_an,
                    const float* __restrict__ w_qn,
                    const float* __restrict__ b_qn,
                    const float* __restrict__ w_qan,
                    const float* __restrict__ b_qan,
                    float* __restrict__ out)
{
    __shared__ float s_logits[PHN * LSTR];   // 50176 B
    __shared__ float s_nlog[NPG];
    __shared__ float s_ha[NPG];
    __shared__ float s_spq[NPG];
    __shared__ float s_qn[NPG];
    __shared__ float s_red[16];
    __shared__ float s_b[2];                 // [0]=group max, [1]=logZ
    __shared__ float s_selLpA;

    const int g    = blockIdx.x;
    const int tid  = threadIdx.x;            // 0..255
    const int lane = tid & 31;
    const int wave = tid >> 5;               // 0..7

    const int act      = a_sel[2 * g];
    const int selLocal = a_sel[2 * g + 1] - g * NPG;   // 0..511

    // ---- per-lane B-matrix base pointers + biases for the 3 N-tiles ----
    const int clo = lane & 15;
    const int kk  = (lane >> 4) << 1;        // 0 or 2

    const float* bp[3];
    float bias[3];
    #pragma unroll
    for (int nt = 0; nt < 3; ++nt) {
        const int col = nt * 16 + clo;
        const float* p;
        float bb;
        if (col == 0)            { p = w_node;                    bb = 0.f; }
        else if (col <= 16)      { p = w_an  + (col - 1)  * D;    bb = b_an[col - 1]; }
        else if (col == 17)      { p = w_qn;                      bb = b_qn[0]; }
        else if (col <= 33)      { p = w_qan + (col - 18) * D;    bb = b_qan[col - 18]; }
        else                     { p = w_node;                    bb = 0.f; }  // padding, unused
        bp[nt]   = p + kk;
        bias[nt] = bb;
    }

    for (int ph = 0; ph < 2; ++ph) {
        // ================= GEMM phase: 256 nodes x 48 cols via WMMA =================
        #pragma unroll
        for (int t = 0; t < 2; ++t) {
            const int mtile = wave * 2 + t;                     // 0..15
            const int row   = g * NPG + ph * PHN + mtile * 16 + clo;
            const float* ap = h + (size_t)row * D + kk;

            v8f c0 = {}, c1 = {}, c2 = {};
            #pragma unroll 8
            for (int k = 0; k < D; k += 4) {
                const float2 af = *(const float2*)(ap + k);
                const float2 b0 = *(const float2*)(bp[0] + k);
                const float2 b1 = *(const float2*)(bp[1] + k);
                const float2 b2 = *(const float2*)(bp[2] + k);
                v2f av; av[0] = af.x; av[1] = af.y;
                v2f bv0; bv0[0] = b0.x; bv0[1] = b0.y;
                v2f bv1; bv1[0] = b1.x; bv1[1] = b1.y;
                v2f bv2; bv2[0] = b2.x; bv2[1] = b2.y;
                c0 = __builtin_amdgcn_wmma_f32_16x16x4_f32(false, av, false, bv0, (short)0, c0, false, false);
                c1 = __builtin_amdgcn_wmma_f32_16x16x4_f32(false, av, false, bv1, (short)0, c1, false, false);
                c2 = __builtin_amdgcn_wmma_f32_16x16x4_f32(false, av, false, bv2, (short)0, c2, false, false);
            }

            // scatter D-matrix + bias to LDS: lane holds column (nt*16+clo), rows v (+8 for hi half)
            const int rhi = (lane >> 4) << 3;
            #pragma unroll
            for (int v = 0; v < 8; ++v) {
                const int r = mtile * 16 + rhi + v;
                float* dst = &s_logits[r * LSTR];
                dst[clo]      = c0[v] + bias[0];
                dst[clo + 16] = c1[v] + bias[1];
                dst[clo + 32] = c2[v] + bias[2];
            }
        }
        __syncthreads();

        // ================= per-node phase: one node per thread =================
        {
            const int jg = ph * PHN + tid;                    // local node in group
            const size_t gnode = (size_t)g * NPG + jg;
            const unsigned char* mp = amask + gnode * A;
            const float* lj = &s_logits[tid * LSTR];

            float alog[A];
            float amax = -3.4e38f;
            bool nm = false;
            #pragma unroll
            for (int aa = 0; aa < A; ++aa) {
                const bool mk = (mp[aa] != 0);
                if (aa >= 1 && mk) nm = true;
                const float v = mk ? lj[1 + aa] : NEGV;
                alog[aa] = v;
                amax = fmaxf(amax, v);
            }
            float zs = 0.f;
            #pragma unroll
            for (int aa = 0; aa < A; ++aa) zs += expf(alog[aa] - amax);
            const float lzs = logf(zs);

            float ha = 0.f, spq = 0.f;
            #pragma unroll
            for (int aa = 0; aa < A; ++aa) {
                const float lp = alog[aa] - amax - lzs;
                const float p  = expf(lp);
                ha  = fmaf(-p, lp, ha);
                spq = fmaf(p, lj[18 + aa], spq);
            }

            s_nlog[jg] = nm ? lj[0] : NEGV;
            s_ha[jg]   = ha;
            s_spq[jg]  = spq;
            s_qn[jg]   = lj[17];
            if (jg == selLocal) s_selLpA = alog[act] - amax - lzs;
        }
        __syncthreads();
    }

    // ================= group reductions over 512 nodes =================
    // 1) max
    float v = fmaxf(s_nlog[tid], s_nlog[tid + 256]);
    #pragma unroll
    for (int off = 16; off > 0; off >>= 1) v = fmaxf(v, __shfl_xor(v, off, 32));
    if (lane == 0) s_red[wave] = v;
    __syncthreads();
    if (tid == 0) {
        float m = s_red[0];
        #pragma unroll
        for (int i = 1; i < 8; ++i) m = fmaxf(m, s_red[i]);
        s_b[0] = m;
    }
    __syncthreads();
    const float M = s_b[0];

    // 2) sum of exp
    float z = expf(s_nlog[tid] - M) + expf(s_nlog[tid + 256] - M);
    #pragma unroll
    for (int off = 16; off > 0; off >>= 1) z += __shfl_xor(z, off, 32);
    if (lane == 0) s_red[wave] = z;
    __syncthreads();
    if (tid == 0) {
        float s = 0.f;
        #pragma unroll
        for (int i = 0; i < 8; ++i) s += s_red[i];
        s_b[1] = logf(s);
    }
    __syncthreads();
    const float logZ = s_b[1];

    // 3) entropy and value sums
    float ent = 0.f, vs = 0.f;
    #pragma unroll
    for (int half = 0; half < 2; ++half) {
        const int idx = tid + half * 256;
        const float lp = s_nlog[idx] - M - logZ;
        const float p  = expf(lp);
        ent = fmaf(p, s_ha[idx] - lp, ent);
        vs  = fmaf(p, s_qn[idx], vs);
    }
    #pragma unroll
    for (int off = 16; off > 0; off >>= 1) {
        ent += __shfl_xor(ent, off, 32);
        vs  += __shfl_xor(vs,  off, 32);
    }
    if (lane == 0) { s_red[wave] = ent; s_red[8 + wave] = vs; }
    __syncthreads();

    if (tid == 0) {
        float entT = 0.f, vsT = 0.f;
        #pragma unroll
        for (int i = 0; i < 8; ++i) { entT += s_red[i]; vsT += s_red[8 + i]; }
        const float lpn_sel = s_nlog[selLocal] - M - logZ;
        out[g]           = lpn_sel + s_selLpA;           // logprob
        out[G + g]       = entT;                          // entropy
        out[2 * G + g]   = s_spq[selLocal] + vsT;         // value
    }
}

extern "C" void kernel_launch(void* const* d_in, const int* in_sizes, int n_in,
                              void* d_out, int out_size, void* d_ws, size_t ws_size,
                              hipStream_t stream) {
    (void)in_sizes; (void)n_in; (void)d_ws; (void)ws_size; (void)out_size;
    const int*           a_sel  = (const int*)d_in[0];
    const float*         h      = (const float*)d_in[1];
    const unsigned char* amask  = (const unsigned char*)d_in[2];
    // d_in[3] = batch_idx (groups are contiguous runs of NPG), d_in[4] = n_nodes (all NPG) — unused
    const float*         w_node = (const float*)d_in[5];
    const float*         w_an   = (const float*)d_in[6];
    const float*         b_an   = (const float*)d_in[7];
    const float*         w_qn   = (const float*)d_in[8];
    const float*         b_qn   = (const float*)d_in[9];
    const float*         w_qan  = (const float*)d_in[10];
    const float*         b_qan  = (const float*)d_in[11];
    float*               out    = (float*)d_out;

    fused_policy_kernel<<<G, 256, 0, stream>>>(a_sel, h, amask, w_node, w_an, b_an,
                                               w_qn, b_qn, w_qan, b_qan, out);
}